// EquivariantReynoldsWrap_51926154609234
// MI455X (gfx1250) — compile-verified
//
#include <hip/hip_runtime.h>
#include <hip/hip_bf16.h>

// Problem constants from the reference
#define BATCH 8
#define CH    64     // C = COUT = 64
#define HW    4096   // H*W = 64*64
#define GRP   8
#define CGRP  4

typedef float v2f __attribute__((ext_vector_type(2)));
typedef float v8f __attribute__((ext_vector_type(8)));

// ---------------------------------------------------------------------------
// Kernel 1: fold the Reynolds-averaged group action into one 64x64 matrix:
//   W_eff[o][c] = (1/G) * sum_g sum_i sum_j A[g][o%4][i] * Wf[(o/4)*4+i][(c/4)*4+j] * Ainv[g][j][c%4]
// ---------------------------------------------------------------------------
__global__ void build_weff_kernel(const float* __restrict__ A,
                                  const float* __restrict__ Ainv,
                                  const float* __restrict__ Wf,
                                  float* __restrict__ Weff) {
    int idx = blockIdx.x * blockDim.x + threadIdx.x;   // 0 .. 4095
    if (idx >= CH * CH) return;
    int o  = idx >> 6;
    int c  = idx & 63;
    int no = o >> 2, r = o & 3;
    int nc = c >> 2, s = c & 3;

    float acc = 0.0f;
    for (int g = 0; g < GRP; ++g) {
        const float* Ag  = A    + g * (CGRP * CGRP);
        const float* Aig = Ainv + g * (CGRP * CGRP);
        for (int i = 0; i < CGRP; ++i) {
            float inner = 0.0f;
            for (int j = 0; j < CGRP; ++j) {
                inner += Wf[(no * 4 + i) * CH + (nc * 4 + j)] * Aig[j * 4 + s];
            }
            acc += Ag[r * 4 + i] * inner;
        }
    }
    Weff[o * CH + c] = acc * (1.0f / (float)GRP);
}

// ---------------------------------------------------------------------------
// Kernel 2: out[b] (64 x 4096) = W_eff (64x64) @ x[b] (64 x 4096)
// using V_WMMA_F32_16X16X4_F32 (full fp32 tensor op, CDNA5-only).
//
// One wave computes a 16(M=out-channel) x 64(N=spatial) tile:
//   4 f32 accumulators (v8f each), A fragment reused across all 4.
// K loop: 64 channels in steps of 4 -> 16 iterations, 4 WMMAs each.
//
// Fragment layouts (ISA 7.12.2, 32-bit operands, wave32):
//   A 16x4 : lanes 0-15 M=0..15 {V0=K0,V1=K1}; lanes 16-31 {V0=K2,V1=K3}
//   B 4x16 : lanes 0-15 N=0..15 {V0=K0,V1=K1}; lanes 16-31 {V0=K2,V1=K3}
//   C 16x16: lane half selects M+=8; VGPR v = row M=v (+8), col N = lane&15
// ---------------------------------------------------------------------------
__global__ __launch_bounds__(256)
void reynolds_gemm_kernel(const float* __restrict__ x,
                          const float* __restrict__ Weff,
                          float* __restrict__ out) {
    const int wave = (blockIdx.x * 256 + threadIdx.x) >> 5;  // 0 .. 2047
    const int lane = threadIdx.x & 31;

    const int batch   = wave >> 8;        // 8 batches, 256 wave-tiles each
    const int rem     = wave & 255;
    const int m_off   = (rem >> 6) << 4;  // 4 M-tiles of 16 rows
    const int p_off   = (rem & 63) << 6;  // 64 N-supertiles of 64 cols

    const int nl  = lane & 15;            // lane within half (M for A, N for B/C)
    const int khi = lane >> 4;            // which K/M half this lane serves
    const int kb  = khi << 1;             // K sub-offset {0,2}

    const float* __restrict__ xb = x   + (size_t)batch * CH * HW;
    float*       __restrict__ ob = out + (size_t)batch * CH * HW;

    v8f acc0 = {}, acc1 = {}, acc2 = {}, acc3 = {};

    const float* wrow = Weff + (m_off + nl) * CH + kb;     // A: row m, cols k..k+1
    const float* xcol = xb + (size_t)kb * HW + p_off + nl; // B: row k, col p

    #pragma unroll
    for (int k0 = 0; k0 < CH; k0 += 4) {
        v2f a = *(const v2f*)(wrow + k0);                  // contiguous float2

        const float* xk = xcol + (size_t)k0 * HW;
        v2f b0, b1, b2, b3;
        b0.x = xk[0];            b0.y = xk[HW];
        b1.x = xk[16];           b1.y = xk[HW + 16];
        b2.x = xk[32];           b2.y = xk[HW + 32];
        b3.x = xk[48];           b3.y = xk[HW + 48];

        acc0 = __builtin_amdgcn_wmma_f32_16x16x4_f32(false, a, false, b0, (short)0, acc0, false, false);
        acc1 = __builtin_amdgcn_wmma_f32_16x16x4_f32(false, a, false, b1, (short)0, acc1, false, false);
        acc2 = __builtin_amdgcn_wmma_f32_16x16x4_f32(false, a, false, b2, (short)0, acc2, false, false);
        acc3 = __builtin_amdgcn_wmma_f32_16x16x4_f32(false, a, false, b3, (short)0, acc3, false, false);
    }

    // Store D: row M = m_off + v + 8*khi, col N = p_off + t*16 + nl
    #pragma unroll
    for (int v = 0; v < 8; ++v) {
        float* op = ob + (size_t)(m_off + v + 8 * khi) * HW + p_off + nl;
        op[0]  = acc0[v];
        op[16] = acc1[v];
        op[32] = acc2[v];
        op[48] = acc3[v];
    }
}

extern "C" void kernel_launch(void* const* d_in, const int* in_sizes, int n_in,
                              void* d_out, int out_size, void* d_ws, size_t ws_size,
                              hipStream_t stream) {
    const float* x    = (const float*)d_in[0];  // (8, 64, 64, 64)
    const float* A    = (const float*)d_in[1];  // (8, 4, 4)
    const float* Ainv = (const float*)d_in[2];  // (8, 4, 4)
    const float* Wf   = (const float*)d_in[3];  // (64, 64)
    float* out  = (float*)d_out;                // (8, 64, 64, 64)
    float* Weff = (float*)d_ws;                 // 64*64 fp32 = 16 KB scratch

    // 1) Fold group algebra into the effective 64x64 matrix.
    build_weff_kernel<<<(CH * CH + 255) / 256, 256, 0, stream>>>(A, Ainv, Wf, Weff);

    // 2) One memory-bound fp32 WMMA GEMM over the whole tensor.
    //    2048 waves total -> 256 blocks x 256 threads (8 wave32 per block).
    reynolds_gemm_kernel<<<256, 256, 0, stream>>>(x, Weff, out);
}